// GraphConvNetwork_20968030339470
// MI455X (gfx1250) — compile-verified
//
#include <hip/hip_runtime.h>
#include <hip/hip_bf16.h>

// ---------------------------------------------------------------------------
// GCN on MI455X (gfx1250): f32 WMMA for node-feature GEMMs, coalesced
// float4 atomic scatter for edge aggregation (the memory-bound dominant phase).
// ---------------------------------------------------------------------------

#define N_NODES  100000
#define N_EDGES  600000
#define N_GRAPHS 2000
#define CDIM     128     // IN_DIM == MODEL_DIM
#define HIDDEN   256
#define N_CONV   4

#define WSTRIDE  132     // LDS stride: 132 % 64 == 4 -> conflict-free WMMA frag reads
#define TILES_PER_BLOCK 8
#define GEMM_LDS_FLOATS (CDIM*WSTRIDE + TILES_PER_BLOCK*16*WSTRIDE)
#define GEMM_LDS_BYTES  (GEMM_LDS_FLOATS * 4)

typedef float v2f __attribute__((ext_vector_type(2)));
typedef float v8f __attribute__((ext_vector_type(8)));

// ---------------------------------------------------------------------------
// GEMM: Out[N x 128] = A[N x 128] @ W[128 x 128] using V_WMMA_F32_16X16X4_F32.
// Block = 8 waves; each wave computes one 16-row x 128-col output strip.
// W staged transposed in LDS (sW[n*132 + k]) so the B fragment (K, K+1) is one
// ds_load_b64; A tile staged per-wave with the same stride for the A fragment.
// ---------------------------------------------------------------------------
__global__ __launch_bounds__(256) void gemm_wmma_kernel(
    const float* __restrict__ A, const float* __restrict__ W,
    float* __restrict__ Out, int nrows) {
  extern __shared__ float smem[];
  float* sW = smem;                       // 128 * 132
  float* sA = smem + CDIM * WSTRIDE;      // 8 waves * 16 * 132

  const int t    = threadIdx.x;
  const int lane = t & 31;
  const int wv   = t >> 5;

  // Cooperative load of W, transposed: sW[n*WSTRIDE + k] = W[k*128 + n]
  for (int j = 0; j < 16; ++j) {
    int i  = t + j * 256;                 // float4 index, 0..4095
    int k  = i >> 5;                      // W row (K)
    int n4 = (i & 31) << 2;               // W col group (N)
    float4 w = ((const float4*)W)[i];
    sW[(n4 + 0) * WSTRIDE + k] = w.x;
    sW[(n4 + 1) * WSTRIDE + k] = w.y;
    sW[(n4 + 2) * WSTRIDE + k] = w.z;
    sW[(n4 + 3) * WSTRIDE + k] = w.w;
  }
  __syncthreads();

  const int tile = blockIdx.x * TILES_PER_BLOCK + wv;
  const int row0 = tile * 16;
  float* aT = sA + wv * (16 * WSTRIDE);

  // Per-wave A tile load (zero-padded past nrows). Same-wave LDS ordering
  // guarantees visibility; no barrier needed for aT.
  for (int j = 0; j < 16; ++j) {
    int i  = lane + j * 32;               // 0..511 float4 slots (16 rows x 32)
    int r  = i >> 5;
    int c4 = i & 31;
    int node = row0 + r;
    float4 v = make_float4(0.f, 0.f, 0.f, 0.f);
    if (node < nrows) v = ((const float4*)A)[node * 32 + c4];
    *(float4*)&aT[r * WSTRIDE + (c4 << 2)] = v;
  }

  const int m  = lane & 15;               // matrix row/col within fragment
  const int hi = lane >> 4;               // half-wave select (K offset +2)

  v8f acc[8];
  #pragma unroll
  for (int nt = 0; nt < 8; ++nt) acc[nt] = (v8f){};

  const float* aBase = aT + m * WSTRIDE + 2 * hi;
  const float* bBase = sW + 2 * hi;

  for (int k0 = 0; k0 < CDIM; k0 += 4) {
    // A frag: VGPR0 = A[m][k0+2*hi], VGPR1 = A[m][k0+1+2*hi]  (one b64 load)
    v2f a = *(const v2f*)(aBase + k0);
    #pragma unroll
    for (int nt = 0; nt < 8; ++nt) {
      // B frag: VGPR0 = W[k0+2*hi][n0+m], VGPR1 = W[k0+1+2*hi][n0+m]
      v2f b = *(const v2f*)(bBase + (nt * 16 + m) * WSTRIDE + k0);
      acc[nt] = __builtin_amdgcn_wmma_f32_16x16x4_f32(
          false, a, false, b, (short)0, acc[nt], false, false);
    }
  }

  // C/D layout: VGPR r holds M = r + 8*hi, N = n0 + m
  if (row0 < nrows) {
    #pragma unroll
    for (int nt = 0; nt < 8; ++nt) {
      #pragma unroll
      for (int r = 0; r < 8; ++r) {
        int row = row0 + r + 8 * hi;
        if (row < nrows) Out[row * CDIM + nt * 16 + m] = acc[nt][r];
      }
    }
  }
}

// ---------------------------------------------------------------------------
// Degree / normalization (self-loop folded in as deg init = 1)
// ---------------------------------------------------------------------------
__global__ void deg_init_kernel(float* __restrict__ deg) {
  int i = blockIdx.x * blockDim.x + threadIdx.x;
  if (i < N_NODES) deg[i] = 1.0f;          // self-loop
}

__global__ void deg_edge_kernel(const int* __restrict__ ei, float* __restrict__ deg) {
  int e = blockIdx.x * blockDim.x + threadIdx.x;
  if (e < N_EDGES) atomicAdd(&deg[ei[N_EDGES + e]], 1.0f);
}

__global__ void dis_kernel(const float* __restrict__ deg, float* __restrict__ dis) {
  int i = blockIdx.x * blockDim.x + threadIdx.x;
  if (i < N_NODES) dis[i] = rsqrtf(deg[i]); // deg >= 1 always
}

// ---------------------------------------------------------------------------
// agg init with self-loop term: agg[n] = hW[n] * dis[n]^2   (wave per node)
// ---------------------------------------------------------------------------
__global__ void selfagg_kernel(const float* __restrict__ hW,
                               const float* __restrict__ dis,
                               float* __restrict__ agg) {
  int t = blockIdx.x * blockDim.x + threadIdx.x;   // N_NODES*32 threads exactly
  int n = t >> 5, lane = t & 31;
  float s = dis[n]; s *= s;
  float4 v = ((const float4*)hW)[n * 32 + lane];
  float4 o; o.x = v.x * s; o.y = v.y * s; o.z = v.z * s; o.w = v.w * s;
  ((float4*)agg)[n * 32 + lane] = o;
}

// ---------------------------------------------------------------------------
// Edge scatter: agg[dst] += hW[src] * dis[src]*dis[dst]   (wave per edge)
// Coalesced float4 gather of the source row, 4x global_atomic_add_f32.
// ---------------------------------------------------------------------------
__global__ void scatter_kernel(const float* __restrict__ hW,
                               const float* __restrict__ dis,
                               const int* __restrict__ ei,
                               float* __restrict__ agg) {
  int t = blockIdx.x * blockDim.x + threadIdx.x;   // N_EDGES*32 threads exactly
  int e = t >> 5, lane = t & 31;
  int s = ei[e];
  int d = ei[N_EDGES + e];
  float nrm = dis[s] * dis[d];
  float4 v = ((const float4*)hW)[s * 32 + lane];
  float* dst = agg + d * CDIM + (lane << 2);
  atomicAdd(dst + 0, v.x * nrm);
  atomicAdd(dst + 1, v.y * nrm);
  atomicAdd(dst + 2, v.z * nrm);
  atomicAdd(dst + 3, v.w * nrm);
}

// ---------------------------------------------------------------------------
// h = relu(agg + bias)
// ---------------------------------------------------------------------------
__global__ void biasrelu_kernel(const float* __restrict__ agg,
                                const float* __restrict__ bias,
                                float* __restrict__ h) {
  int t = blockIdx.x * blockDim.x + threadIdx.x;
  int n = t >> 5, lane = t & 31;
  float4 a = ((const float4*)agg)[n * 32 + lane];
  float4 b = ((const float4*)bias)[lane];
  float4 o;
  o.x = fmaxf(a.x + b.x, 0.f); o.y = fmaxf(a.y + b.y, 0.f);
  o.z = fmaxf(a.z + b.z, 0.f); o.w = fmaxf(a.w + b.w, 0.f);
  ((float4*)h)[n * 32 + lane] = o;
}

// ---------------------------------------------------------------------------
// Global mean pool (sums + counts via atomics)
// ---------------------------------------------------------------------------
__global__ void zero_kernel(float* __restrict__ p, int n) {
  int i = blockIdx.x * blockDim.x + threadIdx.x;
  if (i < n) p[i] = 0.f;
}

__global__ void pool_kernel(const float* __restrict__ h,
                            const int* __restrict__ batch,
                            float* __restrict__ gsum,
                            float* __restrict__ gcnt) {
  int t = blockIdx.x * blockDim.x + threadIdx.x;
  int n = t >> 5, lane = t & 31;
  int g = batch[n];
  float4 v = ((const float4*)h)[n * 32 + lane];
  float* dst = gsum + g * CDIM + (lane << 2);
  atomicAdd(dst + 0, v.x);
  atomicAdd(dst + 1, v.y);
  atomicAdd(dst + 2, v.z);
  atomicAdd(dst + 3, v.w);
  if (lane == 0) atomicAdd(&gcnt[g], 1.0f);
}

// ---------------------------------------------------------------------------
// MLP head: one block per graph. hidden = relu(gmean@w1 + b1); out = hidden@w2 + b2
// ---------------------------------------------------------------------------
__global__ __launch_bounds__(HIDDEN) void mlp_kernel(
    const float* __restrict__ gsum, const float* __restrict__ gcnt,
    const float* __restrict__ w1, const float* __restrict__ b1,
    const float* __restrict__ w2, const float* __restrict__ b2,
    float* __restrict__ out) {
  __shared__ float gm[CDIM];
  __shared__ float red[HIDDEN];
  int g = blockIdx.x, t = threadIdx.x;
  if (t < CDIM) gm[t] = gsum[g * CDIM + t] / fmaxf(gcnt[g], 1.0f);
  __syncthreads();
  float acc = b1[t];
  #pragma unroll 8
  for (int k = 0; k < CDIM; ++k) acc += gm[k] * w1[k * HIDDEN + t];
  red[t] = fmaxf(acc, 0.f) * w2[t];
  __syncthreads();
  for (int s = HIDDEN / 2; s > 0; s >>= 1) {
    if (t < s) red[t] += red[t + s];
    __syncthreads();
  }
  if (t == 0) out[g] = red[0] + b2[0];
}

// ---------------------------------------------------------------------------
// Host orchestration
// Workspace (floats): h | hW | agg | dis | deg | gsum | gcnt  ~= 156 MB
// ---------------------------------------------------------------------------
extern "C" void kernel_launch(void* const* d_in, const int* in_sizes, int n_in,
                              void* d_out, int out_size, void* d_ws, size_t ws_size,
                              hipStream_t stream) {
  (void)in_sizes; (void)n_in; (void)out_size; (void)ws_size;

  const float* x      = (const float*)d_in[0];
  const float* conv_w = (const float*)d_in[1];
  const float* conv_b = (const float*)d_in[2];
  const float* w1     = (const float*)d_in[3];
  const float* b1     = (const float*)d_in[4];
  const float* w2     = (const float*)d_in[5];
  const float* b2     = (const float*)d_in[6];
  const int*   ei     = (const int*)d_in[7];   // [2, N_EDGES]
  const int*   batch  = (const int*)d_in[8];   // [N_NODES]
  float* out = (float*)d_out;

  const size_t NF = (size_t)N_NODES * CDIM;
  float* ws   = (float*)d_ws;
  float* h    = ws;
  float* hW   = h   + NF;
  float* agg  = hW  + NF;
  float* dis  = agg + NF;
  float* deg  = dis + N_NODES;
  float* gsum = deg + N_NODES;
  float* gcnt = gsum + (size_t)N_GRAPHS * CDIM;

  // degree + symmetric normalization (self-loops folded in)
  deg_init_kernel<<<(N_NODES + 255) / 256, 256, 0, stream>>>(deg);
  deg_edge_kernel<<<(N_EDGES + 255) / 256, 256, 0, stream>>>(ei, deg);
  dis_kernel<<<(N_NODES + 255) / 256, 256, 0, stream>>>(deg, dis);

  hipFuncSetAttribute((const void*)gemm_wmma_kernel,
                      hipFuncAttributeMaxDynamicSharedMemorySize, GEMM_LDS_BYTES);

  const int nTiles = (N_NODES + 15) / 16;                        // 6250
  const int gemmBlocks = (nTiles + TILES_PER_BLOCK - 1) / TILES_PER_BLOCK; // 782
  const int nodeWaveBlocks = N_NODES * 32 / 256;                 // 12500
  const int edgeWaveBlocks = N_EDGES * 32 / 256;                 // 75000

  const float* hin = x;
  for (int l = 0; l < N_CONV; ++l) {
    gemm_wmma_kernel<<<gemmBlocks, 256, GEMM_LDS_BYTES, stream>>>(
        hin, conv_w + (size_t)l * CDIM * CDIM, hW, N_NODES);
    selfagg_kernel<<<nodeWaveBlocks, 256, 0, stream>>>(hW, dis, agg);
    scatter_kernel<<<edgeWaveBlocks, 256, 0, stream>>>(hW, dis, ei, agg);
    biasrelu_kernel<<<nodeWaveBlocks, 256, 0, stream>>>(agg, conv_b + (size_t)l * CDIM, h);
    hin = h;
  }

  zero_kernel<<<(N_GRAPHS * CDIM + N_GRAPHS + 255) / 256, 256, 0, stream>>>(
      gsum, N_GRAPHS * CDIM + N_GRAPHS);
  pool_kernel<<<nodeWaveBlocks, 256, 0, stream>>>(h, batch, gsum, gcnt);
  mlp_kernel<<<N_GRAPHS, HIDDEN, 0, stream>>>(gsum, gcnt, w1, b1, w2, b2, out);
}